// Post_process_deconv_80642305950132
// MI455X (gfx1250) — compile-verified
//
#include <hip/hip_runtime.h>
#include <hip/hip_bf16.h>
#include <stdint.h>

namespace {
constexpr int B_ = 2, H_ = 352, W_ = 1216, K_ = 9;
constexpr int TW = 32, TH = 8, HALO = 8;
constexpr int WINW = TW + 2 * HALO;   // 48
constexpr int WINH = TH + 2 * HALO;   // 24
}

typedef __attribute__((ext_vector_type(4))) unsigned int tdm_g0_t;
typedef __attribute__((ext_vector_type(8))) int          tdm_g1_t;
typedef __attribute__((ext_vector_type(4))) int          tdm_g2_t;
typedef __attribute__((ext_vector_type(8))) int          tdm_gx_t;

__device__ __forceinline__ float fetch_depth(int y, int x,
                                             const float* __restrict__ dB,
                                             const float* tile,
                                             int wy0, int wx0) {
  const int ly = y - wy0, lx = x - wx0;
  if ((unsigned)ly < (unsigned)WINH && (unsigned)lx < (unsigned)WINW)
    return tile[ly * WINW + lx];                       // LDS hit (common case)
  if ((unsigned)y < (unsigned)H_ && (unsigned)x < (unsigned)W_)
    return dB[(size_t)y * W_ + x];                     // rare global fallback (L2-resident)
  return 0.0f;                                         // zero-padded out-of-image
}

__global__ __launch_bounds__(TW * TH)
void dcn_post_kernel(const float* __restrict__ depth,
                     const float* __restrict__ weight,
                     const float* __restrict__ offset,
                     const float* __restrict__ wker,
                     const float* __restrict__ bias,
                     float* __restrict__ out) {
  __shared__ float tile[WINH * WINW];

  const int b  = blockIdx.z;
  const int r0 = blockIdx.y * TH;
  const int c0 = blockIdx.x * TW;

  int wy0 = r0 - HALO; wy0 = wy0 < 0 ? 0 : wy0; if (wy0 > H_ - WINH) wy0 = H_ - WINH;
  int wx0 = c0 - HALO; wx0 = wx0 < 0 ? 0 : wx0; if (wx0 > W_ - WINW) wx0 = W_ - WINW;

  const size_t hw = (size_t)H_ * W_;
  const float* dB = depth + (size_t)b * hw;

#if __has_builtin(__builtin_amdgcn_tensor_load_to_lds)
  const bool wave0 = (threadIdx.y == 0);
  if (wave0) {  // wave 0 only: TDM ignores EXEC, avoid duplicate DMAs
    const uint64_t ga  = (uint64_t)(uintptr_t)(dB + (size_t)wy0 * W_ + wx0);
    const uint32_t lds = (uint32_t)(uintptr_t)(&tile[0]);
    tdm_g0_t g0;
    g0.x = 1u;                                              // count=1 (valid D#)
    g0.y = lds;                                             // lds_addr
    g0.z = (uint32_t)ga;                                    // global_addr[31:0]
    g0.w = (uint32_t)((ga >> 32) & 0x01FFFFFFu) | (2u << 30); // addr[56:32] | type=2
    tdm_g1_t g1;
    g1[0] = 0x00020000;                                     // data_size=2 -> 4 bytes
    g1[1] = (W_ & 0xFFFF) << 16;                            // tensor_dim0 lo16
    g1[2] = (W_ >> 16) | ((H_ & 0xFFFF) << 16);             // dim0 hi | dim1 lo
    g1[3] = (H_ >> 16) | (WINW << 16);                      // dim1 hi | tile_dim0
    g1[4] = WINH;                                           // tile_dim1 (tile_dim2=0)
    g1[5] = W_;                                             // tensor_dim0_stride lo
    g1[6] = 0;
    g1[7] = 0;
    tdm_g2_t gz4 = {0, 0, 0, 0};
    tdm_gx_t gz8 = {0, 0, 0, 0, 0, 0, 0, 0};
    __builtin_amdgcn_tensor_load_to_lds(g0, g1, gz4, gz4, gz8, 0);
    // NOTE: no wait here — overlap the DMA with the streaming loads below.
  }
#endif

  const int h = r0 + (int)threadIdx.y;
  const int w = c0 + (int)threadIdx.x;

  // ---- Issue the bandwidth-dominant streaming loads while the TDM DMA runs ----
  const float* wgt = weight + (size_t)b * K_ * hw      + (size_t)h * W_ + w;
  const float* off = offset + (size_t)b * 2 * K_ * hw  + (size_t)h * W_ + w;

  float mk[K_], dy[K_], dx[K_];
#pragma unroll
  for (int k = 0; k < K_; ++k) {
    mk[k] = wgt[(size_t)k * hw];
    dy[k] = off[(size_t)(2 * k)     * hw];
    dx[k] = off[(size_t)(2 * k + 1) * hw];
  }
  const float bs = bias[0];

#if __has_builtin(__builtin_amdgcn_tensor_load_to_lds)
  if (wave0) {
#if __has_builtin(__builtin_amdgcn_s_wait_tensorcnt)
    __builtin_amdgcn_s_wait_tensorcnt(0);
#else
    asm volatile("s_wait_tensorcnt 0x0" ::: "memory");
#endif
  }
#else
  // Cooperative fallback (also keeps the host-side parse clean)
  {
    const int tid = threadIdx.y * TW + threadIdx.x;
    for (int i = tid; i < WINH * WINW; i += TW * TH)
      tile[i] = dB[(size_t)(wy0 + i / WINW) * W_ + (wx0 + i % WINW)];
  }
#endif
  __syncthreads();

  const float dcen = tile[(h - wy0) * WINW + (w - wx0)];

  float Sm = 0.0f, S1 = 0.0f, S2 = 0.0f;
#pragma unroll
  for (int k = 0; k < K_; ++k) {
    const float py = (float)h + (float)(k / 3 - 1) + dy[k];
    const float px = (float)w + (float)(k % 3 - 1) + dx[k];

    const float y0f = floorf(py), x0f = floorf(px);
    const float fy = py - y0f,    fx = px - x0f;
    const int   iy = (int)y0f,    ix = (int)x0f;

    const float v00 = fetch_depth(iy,     ix,     dB, tile, wy0, wx0);
    const float v01 = fetch_depth(iy,     ix + 1, dB, tile, wy0, wx0);
    const float v10 = fetch_depth(iy + 1, ix,     dB, tile, wy0, wx0);
    const float v11 = fetch_depth(iy + 1, ix + 1, dB, tile, wy0, wx0);

    const float val = (v00 * (1.0f - fx) + v01 * fx) * (1.0f - fy)
                    + (v10 * (1.0f - fx) + v11 * fx) * fy;

    const float wk = wker[k];
    Sm += mk[k];
    S1 = fmaf(wk * mk[k], val, S1);
    S2 = fmaf(wk,         val, S2);
  }

  const float res = S1 - (Sm * (1.0f / 9.0f)) * S2 + bs + dcen;
  out[(size_t)b * hw + (size_t)h * W_ + w] = res;
}

extern "C" void kernel_launch(void* const* d_in, const int* in_sizes, int n_in,
                              void* d_out, int out_size, void* d_ws, size_t ws_size,
                              hipStream_t stream) {
  const float* depth  = (const float*)d_in[0];
  const float* weight = (const float*)d_in[1];
  const float* offset = (const float*)d_in[2];
  const float* wker   = (const float*)d_in[3];
  const float* bias   = (const float*)d_in[4];
  float* out = (float*)d_out;

  dim3 block(TW, TH, 1);
  dim3 grid(W_ / TW, H_ / TH, B_);
  dcn_post_kernel<<<grid, block, 0, stream>>>(depth, weight, offset, wker, bias, out);
}